// ContrastiveLossWithAttention_82214263980171
// MI455X (gfx1250) — compile-verified
//
#include <hip/hip_runtime.h>
#include <math.h>

typedef float v2f __attribute__((ext_vector_type(2)));
typedef float v8f __attribute__((ext_vector_type(8)));

__device__ __forceinline__ float clamp01(float x) {
    return fminf(fmaxf(x, 0.0f), 1.0f);
}

// ---------------------------------------------------------------------------
// Kernel A: exact fp32 column sums via V_WMMA_F32_16X16X4_F32.
//   col_gt[b,j]    = sum_i clamp(pred[b,i,j]) * gt[b,i,j]
//   col_avail[b,j] = sum_i gt[b,i,j]
// One wave (32 lanes, EXEC all ones) per (batch, 16-column tile).
// A-matrix = ones(16x4) so D[m,n] = sum_k B[k,n]; result invariant to the
// K<->lane mapping of B as long as each of the 4 rows is loaded once.
// D layout: VGPR0, lanes 0-15 hold (M=0, N=lane) -> lane<16 writes column sums.
// ---------------------------------------------------------------------------
__global__ void __launch_bounds__(32)
colsum_wmma_kernel(const float* __restrict__ pred,
                   const float* __restrict__ gt,
                   float* __restrict__ col_gt,
                   float* __restrict__ col_avail,
                   int N, int M) {
    const int b    = blockIdx.x;
    const int tile = blockIdx.y;
    const int lane = threadIdx.x;          // 0..31
    const int n    = lane & 15;            // column within tile
    const int h    = lane >> 4;            // lane half selects row pair
    const int j    = tile * 16 + n;

    const float* pb = pred + (size_t)b * N * M;
    const float* gb = gt   + (size_t)b * N * M;

    v2f a; a.x = 1.0f; a.y = 1.0f;         // ones A-matrix (16x4 fp32, 2 VGPRs)
    v8f accg = {};                         // C/D accumulators (16x16 fp32)
    v8f accv = {};

    for (int i = 0; i < N; i += 4) {
        const int    r0 = i + 2 * h;       // rows r0, r0+1 for this lane half
        const size_t o0 = (size_t)r0 * M + j;
        const float g0 = gb[o0];
        const float g1 = gb[o0 + M];
        const float p0 = clamp01(pb[o0]);
        const float p1 = clamp01(pb[o0 + M]);

        v2f bg; bg.x = p0 * g0; bg.y = p1 * g1;   // gtp slab (4x16)
        v2f bv; bv.x = g0;      bv.y = g1;        // gt_perm slab (4x16)

        accg = __builtin_amdgcn_wmma_f32_16x16x4_f32(
            false, a, false, bg, (short)0, accg, false, false);
        accv = __builtin_amdgcn_wmma_f32_16x16x4_f32(
            false, a, false, bv, (short)0, accv, false, false);
    }

    if (lane < 16) {                       // row M=0 of D = column sums
        col_gt[(size_t)b * M + j]    = accg[0];
        col_avail[(size_t)b * M + j] = accv[0];
    }
}

// ---------------------------------------------------------------------------
// Kernel B: per-row stats + src_neg. One block (M threads) per (batch,row).
//   row_gt  = sum_j gtp ; row_avail = sum_j gt ; src_pos = sum_j gtp^2
//   tgt_idx = argmax_j gt (first occurrence)
//   src_neg = sum_j [ ((p >= row_gt - beta)*row_avail - gt) * p ]^2  (col-masked)
// Invalid rows (i >= src_ns[b]) are skipped entirely (their loss is masked).
// ---------------------------------------------------------------------------
__global__ void rowstats_kernel(const float* __restrict__ pred,
                                const float* __restrict__ gt,
                                const int*   __restrict__ src_ns,
                                const int*   __restrict__ tgt_ns,
                                const float* __restrict__ beta_p,
                                float* __restrict__ src_pos,
                                float* __restrict__ src_neg,
                                int*   __restrict__ tgt_idx,
                                int N, int M) {
    const int b = blockIdx.y;
    const int i = blockIdx.x;
    const int sn = src_ns[b];
    if (i >= sn) return;                   // uniform for the whole block

    const int   tn   = tgt_ns[b];
    const float beta = *beta_p;
    const int   t    = threadIdx.x;        // 0..M-1

    const size_t row = (size_t)(b * N + i) * M;
    const float p = clamp01(pred[row + t]);
    const float g = gt[row + t];
    const float gp = p * g;

    __shared__ float s0[1024];             // sum gtp   -> then sum att^2
    __shared__ float s1[1024];             // sum gt
    __shared__ float s2[1024];             // sum gtp^2
    __shared__ float s3[1024];             // argmax value
    __shared__ int   si[1024];             // argmax index

    s0[t] = gp;
    s1[t] = g;
    s2[t] = gp * gp;
    s3[t] = g;
    si[t] = t;

    for (int st = blockDim.x >> 1; st > 0; st >>= 1) {
        __syncthreads();
        if (t < st) {
            s0[t] += s0[t + st];
            s1[t] += s1[t + st];
            s2[t] += s2[t + st];
            const float vo = s3[t + st];
            const int   io = si[t + st];
            if (vo > s3[t] || (vo == s3[t] && io < si[t])) {
                s3[t] = vo; si[t] = io;    // first-occurrence argmax
            }
        }
    }
    __syncthreads();
    const float rg   = s0[0];
    const float ra   = s1[0];
    const float pos  = s2[0];
    const int   amax = si[0];
    __syncthreads();                       // everyone read s0[0] before reuse

    float att = (p >= rg - beta ? 1.0f : 0.0f) * ra - g;
    att *= p;
    if (t >= tn) att = 0.0f;               // column mask (row is valid)
    s0[t] = att * att;

    for (int st = blockDim.x >> 1; st > 0; st >>= 1) {
        __syncthreads();
        if (t < st) s0[t] += s0[t + st];
    }
    __syncthreads();
    if (t == 0) {
        const size_t o = (size_t)b * N + i;
        src_pos[o] = pos;
        src_neg[o] = s0[0];
        tgt_idx[o] = amax;
    }
}

// ---------------------------------------------------------------------------
// Kernel C: tgt_neg[b,j] = sum_{i<src_ns} [((p>=col_gt[j]-beta)*col_avail[j]
//                                           - gt) * p]^2   (column-masked)
// One block (M threads) per batch; thread-per-column register accumulation;
// fully coalesced row-major streaming with prefetch hints.
// ---------------------------------------------------------------------------
__global__ void tgtneg_kernel(const float* __restrict__ pred,
                              const float* __restrict__ gt,
                              const float* __restrict__ col_gt,
                              const float* __restrict__ col_avail,
                              const int*   __restrict__ src_ns,
                              const int*   __restrict__ tgt_ns,
                              const float* __restrict__ beta_p,
                              float* __restrict__ tgt_neg,
                              int N, int M) {
    const int b = blockIdx.x;
    const int t = threadIdx.x;             // column
    const int sn = src_ns[b];
    const int tn = tgt_ns[b];
    const float beta = *beta_p;

    const float cav = col_avail[(size_t)b * M + t];
    const float thr = col_gt[(size_t)b * M + t] - beta;

    const float* pb = pred + (size_t)b * N * M;
    const float* gb = gt   + (size_t)b * N * M;

    float acc = 0.0f;
    for (int i = 0; i < sn; ++i) {         // rows >= sn are masked -> skip
        const size_t o = (size_t)i * M + t;
        if (i + 4 < sn) {                  // stream-ahead hint (global_prefetch)
            __builtin_prefetch(&pb[o + (size_t)4 * M], 0, 0);
            __builtin_prefetch(&gb[o + (size_t)4 * M], 0, 0);
        }
        const float p = clamp01(pb[o]);
        const float g = gb[o];
        float att = (p >= thr ? 1.0f : 0.0f) * cav - g;
        att *= p;
        acc += att * att;
    }
    if (t >= tn) acc = 0.0f;               // column mask
    tgt_neg[(size_t)b * M + t] = acc;
}

// ---------------------------------------------------------------------------
// Kernel D: per-batch loss = sum over valid rows of
//   -0.5 * log( src_pos / (1 + src_neg + tgt_neg[tgt_idx]) )
// ---------------------------------------------------------------------------
__global__ void batchloss_kernel(const float* __restrict__ src_pos,
                                 const float* __restrict__ src_neg,
                                 const int*   __restrict__ tgt_idx,
                                 const float* __restrict__ tgt_neg,
                                 const int*   __restrict__ src_ns,
                                 float* __restrict__ batch_loss,
                                 int N, int M) {
    const int b = blockIdx.x;
    const int i = threadIdx.x;             // row
    const int sn = src_ns[b];

    float ell = 0.0f;
    if (i < sn) {
        const size_t o  = (size_t)b * N + i;
        const float pos = src_pos[o];
        const float neg = src_neg[o];
        const float ctn = tgt_neg[(size_t)b * M + tgt_idx[o]];
        const float prob = pos / (1.0f + neg + ctn);
        ell = -0.5f * logf(prob);
    }

    __shared__ float s[1024];
    s[i] = ell;
    for (int st = blockDim.x >> 1; st > 0; st >>= 1) {
        __syncthreads();
        if (i < st) s[i] += s[i + st];
    }
    if (i == 0) batch_loss[b] = s[0];
}

// ---------------------------------------------------------------------------
// Kernel E: final scalar = sum(batch_loss) / sum(src_ns)
// ---------------------------------------------------------------------------
__global__ void finalize_kernel(const float* __restrict__ batch_loss,
                                const int*   __restrict__ src_ns,
                                float* __restrict__ out,
                                int B) {
    const int t = threadIdx.x;
    __shared__ float sl[256];
    __shared__ float sc[256];
    float l = 0.0f, n = 0.0f;
    for (int i = t; i < B; i += blockDim.x) {
        l += batch_loss[i];
        n += (float)src_ns[i];
    }
    sl[t] = l;
    sc[t] = n;
    for (int st = blockDim.x >> 1; st > 0; st >>= 1) {
        __syncthreads();
        if (t < st) { sl[t] += sl[t + st]; sc[t] += sc[t + st]; }
    }
    if (t == 0) out[0] = sl[0] / sc[0];
}

// ---------------------------------------------------------------------------
extern "C" void kernel_launch(void* const* d_in, const int* in_sizes, int n_in,
                              void* d_out, int out_size, void* d_ws, size_t ws_size,
                              hipStream_t stream) {
    const float* pred   = (const float*)d_in[0];
    const float* gtp    = (const float*)d_in[1];
    const int*   src_ns = (const int*)d_in[2];
    const int*   tgt_ns = (const int*)d_in[3];
    const float* beta   = (const float*)d_in[4];
    float*       out    = (float*)d_out;

    const int B  = in_sizes[2];                    // 64
    const long long NM = (long long)in_sizes[0] / B;
    int N = 1;
    while ((long long)N * N < NM) ++N;             // N == M (512)
    const int M = N;

    // Workspace carve-up (fp32 unless noted)
    float* col_gt     = (float*)d_ws;                      // B*M
    float* col_avail  = col_gt     + (size_t)B * M;        // B*M
    float* src_pos    = col_avail  + (size_t)B * M;        // B*N
    float* src_neg    = src_pos    + (size_t)B * N;        // B*N
    int*   tgt_idx    = (int*)(src_neg + (size_t)B * N);   // B*N (int)
    float* tgt_neg    = (float*)(tgt_idx + (size_t)B * N); // B*M
    float* batch_loss = tgt_neg    + (size_t)B * M;        // B

    // A: fp32 WMMA column sums — one wave per (batch, 16-col tile)
    colsum_wmma_kernel<<<dim3(B, M / 16), 32, 0, stream>>>(
        pred, gtp, col_gt, col_avail, N, M);

    // B: per-row stats + src_neg — one block per (row, batch)
    rowstats_kernel<<<dim3(N, B), M, 0, stream>>>(
        pred, gtp, src_ns, tgt_ns, beta, src_pos, src_neg, tgt_idx, N, M);

    // C: tgt_neg — one block per batch, thread per column
    tgtneg_kernel<<<B, M, 0, stream>>>(
        pred, gtp, col_gt, col_avail, src_ns, tgt_ns, beta, tgt_neg, N, M);

    // D: per-batch loss
    batchloss_kernel<<<B, N, 0, stream>>>(
        src_pos, src_neg, tgt_idx, tgt_neg, src_ns, batch_loss, N, M);

    // E: final scalar
    finalize_kernel<<<1, 64, 0, stream>>>(batch_loss, src_ns, out, B);
}